// GaussianKernel_62156766708116
// MI455X (gfx1250) — compile-verified
//
#include <hip/hip_runtime.h>

typedef __attribute__((ext_vector_type(2))) float v2f;
typedef __attribute__((ext_vector_type(8))) float v8f;

#define NTAP 11
#define PAD 5
#define KCHUNKS 7   // 28 K-values cover the 26-wide input window (last 2 rows have zero weight)

// Problem geometry (compile-time): x is (3,1,160,192,224) fp32, NCDHW.
#define NN 3
#define DD 160
#define HD 192
#define WD 224
#define WTILES (WD / 16)                 // 14
constexpr unsigned S1 = (unsigned)DD * HD * WD;   // batch stride = 6881280

// ---------------------------------------------------------------------------
// Prep: recover separable 1-D kernel from dense 11x11x11 (k1[i] = sum dense[i,:,:],
// since sum(k1)=1), then bake per-(chunk,lane) WMMA weight fragments into a
// 7 x 32 x float2 table. The same fragment works as the A operand (16x4 f32:
// lanes0-15 M=lane%16, V0=K, V1=K+1; lanes16-31 K+=2) and as the B operand
// (4x16 f32: same pattern with N=lane%16): tap index = Kglobal - lane%16.
// ---------------------------------------------------------------------------
__global__ void gauss3d_prep(const float* __restrict__ dense, float* __restrict__ wtab) {
    __shared__ float gs[NTAP];
    const int t = threadIdx.x;
    if (t < NTAP) {
        const float* p = dense + t * NTAP * NTAP;
        float s = 0.0f;
        for (int i = 0; i < NTAP * NTAP; ++i) s += p[i];
        gs[t] = s;
    }
    __syncthreads();
    if (t < KCHUNKS * 32) {
        const int j    = t >> 5;
        const int lane = t & 31;
        const int k    = 4 * j + 2 * (lane >> 4);
        const int p    = lane & 15;
        const int t0   = k - p;
        const int t1   = k + 1 - p;
        wtab[2 * t]     = (t0 >= 0 && t0 < NTAP) ? gs[t0] : 0.0f;
        wtab[2 * t + 1] = (t1 >= 0 && t1 < NTAP) ? gs[t1] : 0.0f;
    }
}

// ---------------------------------------------------------------------------
// Pass along the contiguous (w) axis. One wave per 16-row x 16-w tile.
// A = data (M = row, K = window pos), B = banded weights, D[M,N] = out[row M, w0+N].
// All addressing is 32-bit element offsets off the uniform kernel-arg pointers.
// ---------------------------------------------------------------------------
__global__ __launch_bounds__(256) void gauss3d_conv_w(const float* __restrict__ in,
                                                      float* __restrict__ out,
                                                      const float* __restrict__ wtab) {
    const int lane = threadIdx.x & 31;
    const int tile = blockIdx.x * 8 + (threadIdx.x >> 5);
    const int w0   = (tile % WTILES) * 16;
    const int r0   = (tile / WTILES) * 16;

    const v2f* wt = (const v2f*)wtab;
    v2f wf[KCHUNKS];
#pragma unroll
    for (int j = 0; j < KCHUNKS; ++j) wf[j] = wt[j * 32 + lane];

    const unsigned rowOff = (unsigned)(r0 + (lane & 15)) * WD;
    const int kBase = w0 - PAD + 2 * (lane >> 4);

    v8f acc = {};
#pragma unroll
    for (int j = 0; j < KCHUNKS; ++j) {
        const int x0 = kBase + 4 * j;
        const int x1 = x0 + 1;
        const int c0 = min(max(x0, 0), WD - 1);
        const int c1 = min(max(x1, 0), WD - 1);
        float f0 = in[rowOff + (unsigned)c0];
        float f1 = in[rowOff + (unsigned)c1];
        v2f a;
        a.x = (x0 == c0) ? f0 : 0.0f;
        a.y = (x1 == c1) ? f1 : 0.0f;
        acc = __builtin_amdgcn_wmma_f32_16x16x4_f32(false, a, false, wf[j],
                                                    (short)0, acc, false, false);
    }

    // D layout: VGPR v -> M = v + 8*(lane/16), N = lane%16
    unsigned o = (unsigned)(r0 + (lane >> 4) * 8) * WD + (unsigned)(w0 + (lane & 15));
#pragma unroll
    for (int v = 0; v < 8; ++v) {
        out[o] = acc[v];
        o += WD;
    }
}

// ---------------------------------------------------------------------------
// Pass along a strided axis (h or d), fully specialized at compile time.
// One wave per 16-w x 16-line tile.
// A = banded weights, B = data (K = window pos, N = w), D[M,N] = out[c0+M, w0+N].
// ---------------------------------------------------------------------------
template <int STRIDE, int EXTENT, int CTILES, int O2, int S2>
__global__ __launch_bounds__(256) void gauss3d_conv_line(const float* __restrict__ in,
                                                         float* __restrict__ out,
                                                         const float* __restrict__ wtab) {
    const int lane = threadIdx.x & 31;
    const int tile = blockIdx.x * 8 + (threadIdx.x >> 5);
    const int wT   = tile % WTILES;
    int rest       = tile / WTILES;
    const int cT   = rest % CTILES;
    rest           = rest / CTILES;
    const int o2   = rest % O2;
    const int o1   = rest / O2;

    const unsigned base = (unsigned)o1 * S1 + (unsigned)o2 * (unsigned)S2;
    const int w0 = wT * 16;
    const int c0 = cT * 16;

    const v2f* wt = (const v2f*)wtab;
    v2f wf[KCHUNKS];
#pragma unroll
    for (int j = 0; j < KCHUNKS; ++j) wf[j] = wt[j * 32 + lane];

    const unsigned bw   = base + (unsigned)(w0 + (lane & 15));
    const int      kBase = c0 - PAD + 2 * (lane >> 4);

    v8f acc = {};
#pragma unroll
    for (int j = 0; j < KCHUNKS; ++j) {
        const int y0 = kBase + 4 * j;
        const int y1 = y0 + 1;
        const int p0 = min(max(y0, 0), EXTENT - 1);
        const int p1 = min(max(y1, 0), EXTENT - 1);
        float f0 = in[bw + (unsigned)p0 * STRIDE];
        float f1 = in[bw + (unsigned)p1 * STRIDE];
        v2f b;
        b.x = (y0 == p0) ? f0 : 0.0f;
        b.y = (y1 == p1) ? f1 : 0.0f;
        acc = __builtin_amdgcn_wmma_f32_16x16x4_f32(false, wf[j], false, b,
                                                    (short)0, acc, false, false);
    }

    // D layout: VGPR v -> M = v + 8*(lane/16), N = lane%16
    unsigned o = base + (unsigned)(c0 + (lane >> 4) * 8) * STRIDE
                      + (unsigned)(w0 + (lane & 15));
#pragma unroll
    for (int v = 0; v < 8; ++v) {
        out[o] = acc[v];
        o += STRIDE;
    }
}

extern "C" void kernel_launch(void* const* d_in, const int* in_sizes, int n_in,
                              void* d_out, int out_size, void* d_ws, size_t ws_size,
                              hipStream_t stream) {
    (void)in_sizes; (void)n_in; (void)out_size; (void)ws_size;

    const float* x     = (const float*)d_in[0];
    const float* dense = (const float*)d_in[1];
    float*       out   = (float*)d_out;
    float*       wtab  = (float*)d_ws;                    // 7*32*2 floats = 1792 B
    float*       tmp   = (float*)((char*)d_ws + 4096);    // ping-pong intermediate (~83 MB)

    gauss3d_prep<<<1, 256, 0, stream>>>(dense, wtab);

    const int tiles = NN * DD * (HD / 16) * WTILES;       // 80640 for every pass
    const int blocks = tiles / 8;                         // 10080

    // Pass 1: conv along w, x -> out. 92160 rows of 224 (all multiples of 16).
    gauss3d_conv_w<<<blocks, 256, 0, stream>>>(x, out, wtab);

    // Pass 2: conv along h, out -> tmp. outer=(n,d), stride along h = WD.
    gauss3d_conv_line<WD, HD, HD / 16, DD, HD * WD>
        <<<blocks, 256, 0, stream>>>(out, tmp, wtab);

    // Pass 3: conv along d, tmp -> out. outer=(n,h), stride along d = HD*WD.
    gauss3d_conv_line<HD * WD, DD, DD / 16, HD, WD>
        <<<blocks, 256, 0, stream>>>(tmp, out, wtab);
}